// GATLayer_50242527428896
// MI455X (gfx1250) — compile-verified
//
#include <hip/hip_runtime.h>

#define N_NODES 100000
#define E_EDGES 1600000
#define IN_DIM  128
#define HEADS   8
#define OUT_DIM 16
#define HD      128   // HEADS*OUT_DIM

typedef __attribute__((ext_vector_type(2))) float v2f;
typedef __attribute__((ext_vector_type(8))) float v8f;
typedef unsigned int u32x4 __attribute__((ext_vector_type(4)));
typedef int i32x4 __attribute__((ext_vector_type(4)));
typedef int i32x8 __attribute__((ext_vector_type(8)));

#if defined(__has_builtin)
#if __has_builtin(__builtin_amdgcn_tensor_load_to_lds) && \
    __has_builtin(__builtin_amdgcn_s_wait_tensorcnt)
#define HAVE_TDM 1
#endif
#endif

// ---------------- workspace layout (in floats) ----------------
// wlrP : 64 kpairs x 16 cols x 2 = 2048        (paired [w_l|w_r] for WMMA B-frags)
// h    : N*128
// al   : N*8
// ar   : N*8
// m    : N*8 (stored as ordered-encoded uint)
// den  : N*8
#define WLRP_OFF 0
#define H_OFF    2048
#define AL_OFF   (H_OFF + N_NODES * HD)
#define AR_OFF   (AL_OFF + N_NODES * HEADS)
#define M_OFF    (AR_OFF + N_NODES * HEADS)
#define DEN_OFF  (M_OFF + N_NODES * HEADS)

// Order-preserving float<->uint mapping so float max can use global_atomic_max_u32.
__device__ __forceinline__ unsigned enc_f(float f) {
  unsigned u = __float_as_uint(f);
  return (u >> 31) ? ~u : (u | 0x80000000u);
}
__device__ __forceinline__ float dec_f(unsigned e) {
  unsigned u = (e >> 31) ? (e & 0x7FFFFFFFu) : ~e;
  return __uint_as_float(u);
}

// ---------------------------------------------------------------
// init: zero out[] (N*128) and denom, set m = enc(-inf)
// ---------------------------------------------------------------
__global__ void gat_init(float* __restrict__ out, float* __restrict__ den,
                         unsigned* __restrict__ m) {
  int i = blockIdx.x * blockDim.x + threadIdx.x;
  if (i < N_NODES * HD) out[i] = 0.0f;
  if (i < N_NODES * HEADS) {
    den[i] = 0.0f;
    m[i] = 0x007FFFFFu;  // enc(-inf)
  }
}

// ---------------------------------------------------------------
// Fold attn_l/attn_r into W:  wlrP[kp][col] = {w(2kp,col), w(2kp+1,col)}
// col 0..7 -> w_l head=col ; col 8..15 -> w_r head=col-8
// ---------------------------------------------------------------
__global__ void gat_build_wlr(const float* __restrict__ W,
                              const float* __restrict__ attn_l,
                              const float* __restrict__ attn_r,
                              float* __restrict__ wlrP) {
  int i = blockIdx.x * blockDim.x + threadIdx.x;  // 0..1023
  if (i >= 64 * 16) return;
  int kp = i >> 4, col = i & 15;
  int head = col & 7;
  const float* av = ((col < 8) ? attn_l : attn_r) + head * OUT_DIM;
  float s0 = 0.f, s1 = 0.f;
  const float* w0 = W + (2 * kp) * HD + head * OUT_DIM;
  const float* w1 = W + (2 * kp + 1) * HD + head * OUT_DIM;
#pragma unroll
  for (int d = 0; d < OUT_DIM; ++d) {
    s0 += w0[d] * av[d];
    s1 += w1[d] * av[d];
  }
  wlrP[i * 2 + 0] = s0;
  wlrP[i * 2 + 1] = s1;
}

// ---------------------------------------------------------------
// GEMM: h = x @ W  (f32 WMMA 16x16x4), plus fused a_l/a_r as tile 8.
// W (64 KB, row-major) staged into LDS via the Tensor Data Mover.
// One wave computes a 16-row block across all 8 column tiles + the lr tile.
// K-loop is only partially unrolled so LDS B-fragments are NOT hoisted
// into (spilled) registers across the tile loop.
// ---------------------------------------------------------------
__global__ __launch_bounds__(256)
void gat_gemm(const float* __restrict__ x, const float* __restrict__ W,
              const float* __restrict__ wlrP, float* __restrict__ h,
              float* __restrict__ al, float* __restrict__ ar) {
  __shared__ float WL[128 * 128];  // 64 KB, row-major copy of W

#if HAVE_TDM
  if (threadIdx.x < 32) {  // one TDM descriptor per workgroup (TDM ignores EXEC)
    unsigned lds_off = (unsigned)(unsigned long long)(&WL[0]);
    unsigned long long ga = (unsigned long long)W;
    u32x4 g0;
    g0.x = 1u;                                   // count=1 valid descriptor
    g0.y = lds_off;                              // lds_addr  (bits 63:32)
    g0.z = (unsigned)(ga & 0xFFFFFFFFu);         // global_addr[31:0]
    g0.w = (unsigned)(ga >> 32) | (2u << 30);    // global_addr[56:32] | type=2
    i32x8 g1;
    g1[0] = (int)(2u << 16);                     // data_size = 4B
    g1[1] = (int)(128u << 16);                   // tensor_dim0 = 128 (bits 63:48)
    g1[2] = (int)(128u << 16);                   // tensor_dim1 = 128 (bits 95:80)
    g1[3] = (int)(128u << 16);                   // tile_dim0  = 128 (bits 127:112)
    g1[4] = 128;                                 // tile_dim1  = 128 (bits 143:128)
    g1[5] = 128;                                 // tensor_dim0_stride = 128
    g1[6] = 0;
    g1[7] = 0;
    i32x4 z4 = {0, 0, 0, 0};
#if __clang_major__ >= 23
    i32x8 z8 = {0, 0, 0, 0, 0, 0, 0, 0};
    __builtin_amdgcn_tensor_load_to_lds(g0, g1, z4, z4, z8, 0);
#else
    __builtin_amdgcn_tensor_load_to_lds(g0, g1, z4, z4, 0);
#endif
    __builtin_amdgcn_s_wait_tensorcnt(0);
  }
  __syncthreads();
#else
  for (int i = threadIdx.x; i < 128 * 128; i += 256) WL[i] = W[i];
  __syncthreads();
#endif

  const int wave = threadIdx.x >> 5;
  const int lane = threadIdx.x & 31;
  const int m16 = lane & 15;        // A: row in tile / B,C,D: column
  const int hi = lane >> 4;         // half-wave selector (K pairs; rows M vs M+8)
  const int nTiles = N_NODES / 16;  // 6250 exactly

  for (int tile = blockIdx.x * 8 + wave; tile < nTiles; tile += gridDim.x * 8) {
    const int rowbase = tile * 16;
    const float* xrow = x + (rowbase + m16) * IN_DIM + hi * 2;
    __builtin_prefetch(xrow, 0, 0);  // global_prefetch_b8

    v8f acc[9];
#pragma unroll
    for (int c = 0; c < 9; ++c) acc[c] = (v8f){0, 0, 0, 0, 0, 0, 0, 0};

#pragma unroll 4
    for (int t = 0; t < 32; ++t) {          // K = 4*t .. 4*t+3
      v2f a = *(const v2f*)(xrow + 4 * t);  // x[row, 4t+2hi], x[row, 4t+2hi+1]
      const float* wl0 = WL + (4 * t + 2 * hi) * 128;  // row k, row k+1 at +128
#pragma unroll
      for (int c = 0; c < 8; ++c) {
        v2f b;
        b.x = wl0[c * 16 + m16];        // ds_load_b32
        b.y = wl0[128 + c * 16 + m16];  // ds_load_b32
        acc[c] = __builtin_amdgcn_wmma_f32_16x16x4_f32(
            false, a, false, b, (short)0, acc[c], false, false);
      }
      v2f b8 = *(const v2f*)(wlrP + ((2 * t + hi) * 16 + m16) * 2);
      acc[8] = __builtin_amdgcn_wmma_f32_16x16x4_f32(
          false, a, false, b8, (short)0, acc[8], false, false);
    }

    // store h tile: C/D layout -> lane m16 = col, VGPR v = row (v or v+8)
#pragma unroll
    for (int c = 0; c < 8; ++c)
#pragma unroll
      for (int v = 0; v < 8; ++v)
        h[(rowbase + v + 8 * hi) * HD + c * 16 + m16] = acc[c][v];

    // tile 8: cols 0..7 = a_l heads, cols 8..15 = a_r heads
#pragma unroll
    for (int v = 0; v < 8; ++v) {
      int r = rowbase + v + 8 * hi;
      float val = acc[8][v];
      if (m16 < 8) al[r * HEADS + m16] = val;
      else         ar[r * HEADS + (m16 - 8)] = val;
    }
  }
}

// ---------------------------------------------------------------
// edge pass 1: e = leaky_relu(a_l[src] + a_r[dst]); atomic-max into m[dst]
// ---------------------------------------------------------------
__global__ void gat_edge1(const int* __restrict__ ei, const float* __restrict__ al,
                          const float* __restrict__ ar, float* __restrict__ alphaOut,
                          unsigned* __restrict__ m) {
  int gid = blockIdx.x * blockDim.x + threadIdx.x;
  if (gid >= E_EDGES * HEADS) return;
  int e = gid >> 3, hh = gid & 7;
  int src = ei[e];
  int dst = ei[E_EDGES + e];
  float v = al[src * HEADS + hh] + ar[dst * HEADS + hh];
  v = (v > 0.0f) ? v : 0.2f * v;  // leaky relu
  alphaOut[gid] = v;              // stash raw e in the alpha output slot
  atomicMax(&m[dst * HEADS + hh], enc_f(v));
}

// ---------------------------------------------------------------
// edge pass 2: ex = exp(e - m[dst]); atomic-add into denom[dst]
// ---------------------------------------------------------------
__global__ void gat_edge2(const int* __restrict__ ei, float* __restrict__ alphaOut,
                          const unsigned* __restrict__ m, float* __restrict__ den) {
  int gid = blockIdx.x * blockDim.x + threadIdx.x;
  if (gid >= E_EDGES * HEADS) return;
  int e = gid >> 3, hh = gid & 7;
  int dst = ei[E_EDGES + e];
  float mm = dec_f(m[dst * HEADS + hh]);
  float ex = __expf(alphaOut[gid] - mm);
  alphaOut[gid] = ex;
  atomicAdd(&den[dst * HEADS + hh], ex);
}

// ---------------------------------------------------------------
// edge pass 3: alpha = ex/(den+eps); out[dst] += alpha * h[src]
// 256-thread block handles 2 edges (128 lanes == HD per edge).
// ---------------------------------------------------------------
__global__ __launch_bounds__(256)
void gat_edge3(const int* __restrict__ ei, const float* __restrict__ h,
               const float* __restrict__ den, float* __restrict__ alphaOut,
               float* __restrict__ out) {
  int e = blockIdx.x * 2 + (threadIdx.x >> 7);
  int t = threadIdx.x & 127;          // dim within node feature
  int head = t >> 4;
  int src = ei[e];
  int dst = ei[E_EDGES + e];
  float ex = alphaOut[e * HEADS + head];
  float d = den[dst * HEADS + head];
  float alpha = ex / (d + 1e-12f);
  float msg = alpha * h[src * HD + t];
  __syncthreads();                    // all raw-ex reads done before overwrite
  if ((t & 15) == 0) alphaOut[e * HEADS + head] = alpha;
  atomicAdd(&out[dst * HD + t], msg);
}

// ---------------------------------------------------------------
extern "C" void kernel_launch(void* const* d_in, const int* in_sizes, int n_in,
                              void* d_out, int out_size, void* d_ws, size_t ws_size,
                              hipStream_t stream) {
  const float* x      = (const float*)d_in[0];
  const int*   ei     = (const int*)d_in[1];
  const float* W      = (const float*)d_in[2];
  const float* attn_l = (const float*)d_in[3];
  const float* attn_r = (const float*)d_in[4];

  float* out      = (float*)d_out;                    // [N, 128]
  float* alphaOut = (float*)d_out + N_NODES * HD;     // [E, 8]

  float*    ws   = (float*)d_ws;
  float*    wlrP = ws + WLRP_OFF;
  float*    h    = ws + H_OFF;
  float*    al   = ws + AL_OFF;
  float*    ar   = ws + AR_OFF;
  unsigned* m    = (unsigned*)(ws + M_OFF);
  float*    den  = ws + DEN_OFF;

  hipLaunchKernelGGL(gat_init, dim3((N_NODES * HD + 255) / 256), dim3(256), 0, stream,
                     out, den, m);
  hipLaunchKernelGGL(gat_build_wlr, dim3(4), dim3(256), 0, stream,
                     W, attn_l, attn_r, wlrP);
  hipLaunchKernelGGL(gat_gemm, dim3(160), dim3(256), 0, stream,
                     x, W, wlrP, h, al, ar);
  hipLaunchKernelGGL(gat_edge1, dim3((E_EDGES * HEADS + 255) / 256), dim3(256), 0, stream,
                     ei, al, ar, alphaOut, m);
  hipLaunchKernelGGL(gat_edge2, dim3((E_EDGES * HEADS + 255) / 256), dim3(256), 0, stream,
                     ei, alphaOut, m, den);
  hipLaunchKernelGGL(gat_edge3, dim3(E_EDGES / 2), dim3(256), 0, stream,
                     ei, h, den, alphaOut, out);
}